// InvariantPointAttention_86706799772230
// MI455X (gfx1250) — compile-verified
//
#include <hip/hip_runtime.h>
#include <hip/hip_bf16.h>
#include <math.h>

// ---------------- problem constants ----------------
#define LSEQ 768
#define CS   256
#define CZ   128
#define NH   8
#define CHD  16
#define PQK  4
#define PVN  8
#define OUTD 1408   // NH*CHD + NH*PVN*3 + NH*PVN + NH*CZ = 128+192+64+1024

typedef float v2f __attribute__((ext_vector_type(2)));
typedef float v8f __attribute__((ext_vector_type(8)));

// D = A(16x4) * B(4x16) + C, full fp32 WMMA (wave32)
static __device__ __forceinline__ v8f wmma4(v2f a, v2f b, v8f c) {
    return __builtin_amdgcn_wmma_f32_16x16x4_f32(false, a, false, b, (short)0, c,
                                                 false, false);
}

// -----------------------------------------------------------------------------
// Templated batched GEMM: C[z] = A[z] (MxK, row-major LDA) @ B[z] (KxN, row-major LDB)
// Compile-time strides/K -> pointer-increment addressing, dense WMMA issue.
// WIDE=false: one wave per 16x16 tile.  WIDE=true: one wave per 16x32 tile
// (two accumulators share each A fragment).
// grid = (ceil(M/16), ceil(N/TN), batch)
// -----------------------------------------------------------------------------
template <int LDA, int LDB, int LDC, int K, bool WIDE>
__global__ void __launch_bounds__(32)
gemm_t(const float* __restrict__ A, long long sA,
       const float* __restrict__ B, long long sB,
       float* __restrict__ C, long long sC,
       int M, int N) {
    A += (long long)blockIdx.z * sA;
    B += (long long)blockIdx.z * sB;
    C += (long long)blockIdx.z * sC;
    const int m0 = blockIdx.x * 16;
    const int n0 = blockIdx.y * (WIDE ? 32 : 16);
    const int lane = threadIdx.x & 31;
    const int half = lane >> 4;   // selects K sub-pair {0,1} vs {2,3}
    const int lp   = lane & 15;

    // clamped indices: no EXEC divergence before the WMMA ops
    int r  = m0 + lp;        if (r  >= M) r  = M - 1;
    int c0 = n0 + lp;        if (c0 >= N) c0 = N - 1;
    int c1 = WIDE ? (n0 + 16 + lp) : c0;
    if (c1 >= N) c1 = N - 1;

    const float* Ap  = A + (long long)r * LDA + half * 2;
    const float* Bp0 = B + (long long)(half * 2) * LDB + c0;
    const float* Bp1 = B + (long long)(half * 2) * LDB + c1;

    v8f acc0 = {}, acc1 = {};
#pragma unroll 4
    for (int kb = 0; kb < K; kb += 4) {
        if (kb + 32 < K) {                       // uniform branch; HBM-bound op
            __builtin_prefetch(Ap + 32, 0, 0);
            __builtin_prefetch(Bp0 + 32 * LDB, 0, 0);
        }
        v2f a, b0;
        a.x  = Ap[0];   a.y  = Ap[1];
        b0.x = Bp0[0];  b0.y = Bp0[LDB];
        acc0 = wmma4(a, b0, acc0);
        if (WIDE) {
            v2f b1;
            b1.x = Bp1[0];  b1.y = Bp1[LDB];
            acc1 = wmma4(a, b1, acc1);
        }
        Ap  += 4;
        Bp0 += 4 * LDB;
        Bp1 += 4 * LDB;
    }

    const int cc0 = n0 + lp;
    if (cc0 < N) {
        for (int g = 0; g < 8; ++g) {
            const int row = m0 + g + half * 8;
            if (row < M) C[(long long)row * LDC + cc0] = acc0[g];
        }
    }
    if (WIDE) {
        const int cc1 = n0 + 16 + lp;
        if (cc1 < N) {
            for (int g = 0; g < 8; ++g) {
                const int row = m0 + g + half * 8;
                if (row < M) C[(long long)row * LDC + cc1] = acc1[g];
            }
        }
    }
}

// -----------------------------------------------------------------------------
// Apply frame: p_global = R @ p_local + t  (in place), and sum of squares per (l,h)
// pts layout: [l, h*P*3 + p*3 + c];  one thread per (l,h)
// -----------------------------------------------------------------------------
__global__ void transform_pts(const float* __restrict__ R, const float* __restrict__ t,
                              float* __restrict__ pts, float* __restrict__ sq, int P) {
    int idx = blockIdx.x * blockDim.x + threadIdx.x;   // l*NH + h
    if (idx >= LSEQ * NH) return;
    const int l = idx / NH;
    const float* Rl = R + l * 9;
    const float* tl = t + l * 3;
    float* p = pts + (long long)idx * P * 3;
    float acc = 0.f;
    for (int pp = 0; pp < P; ++pp) {
        const float x = p[0], y = p[1], z = p[2];
        const float gx = Rl[0]*x + Rl[1]*y + Rl[2]*z + tl[0];
        const float gy = Rl[3]*x + Rl[4]*y + Rl[5]*z + tl[1];
        const float gz = Rl[6]*x + Rl[7]*y + Rl[8]*z + tl[2];
        p[0] = gx; p[1] = gy; p[2] = gz; p += 3;
        acc += gx*gx + gy*gy + gz*gz;
    }
    if (sq) sq[idx] = acc;
}

__global__ void gamma_kernel(const float* __restrict__ pw, float* __restrict__ gamma) {
    const int h = threadIdx.x;
    if (h < NH) gamma[h] = log1pf(expf(pw[h]));    // softplus
}

// -----------------------------------------------------------------------------
// Fused logits: per (h, 16x16 tile of i x j), one wave.
//   acc1 = q_i . k_j (K=16),  acc2 = qpts_i . kpts_j (K=12)
//   logit = w_l * (acc1/4 + pair_bias - 0.5*gamma*w_c*max(qsq+ksq-2*acc2, 0))
// Stored (pre-softmax) into attn[h,i,j].
// -----------------------------------------------------------------------------
__global__ void __launch_bounds__(32)
logits_kernel(const float* __restrict__ q,  const float* __restrict__ k,
              const float* __restrict__ qp, const float* __restrict__ kp,
              const float* __restrict__ qsq, const float* __restrict__ ksq,
              const float* __restrict__ pb,  const float* __restrict__ gamma,
              const float* __restrict__ mask, float* __restrict__ attn) {
    const int i0 = blockIdx.x * 16, j0 = blockIdx.y * 16, h = blockIdx.z;
    const int lane = threadIdx.x & 31;
    const int half = lane >> 4, lp = lane & 15;

    v8f acc1 = {}, acc2 = {};
    const float* Aq = q  + h * CHD;        // [i, 128] row-major, head slice
    const float* Bk = k  + h * CHD;
    const float* Ap = qp + h * (PQK * 3);  // [i, 96]
    const float* Bp = kp + h * (PQK * 3);

#pragma unroll
    for (int kb = 0; kb < CHD; kb += 4) {            // scalar q.k, K=16
        const int kk = kb + half * 2;
        v2f a, b;
        const int r = i0 + lp, c = j0 + lp;
        a.x = Aq[r * (NH*CHD) + kk];     a.y = Aq[r * (NH*CHD) + kk + 1];
        b.x = Bk[c * (NH*CHD) + kk];     b.y = Bk[c * (NH*CHD) + kk + 1];
        acc1 = wmma4(a, b, acc1);
    }
#pragma unroll
    for (int kb = 0; kb < PQK * 3; kb += 4) {        // point q.k, K=12
        const int kk = kb + half * 2;
        v2f a, b;
        const int r = i0 + lp, c = j0 + lp;
        a.x = Ap[r * (NH*PQK*3) + kk];   a.y = Ap[r * (NH*PQK*3) + kk + 1];
        b.x = Bp[c * (NH*PQK*3) + kk];   b.y = Bp[c * (NH*PQK*3) + kk + 1];
        acc2 = wmma4(a, b, acc2);
    }

    const float g   = gamma[h];
    const float w_c = 0.23570226039551587f;   // sqrt(2/(9*PQK))
    const float w_l = 0.57735026918962576f;   // sqrt(1/3)
    const int   col = j0 + lp;
    const float ksv = ksq[col * NH + h];
    const float mj  = mask[col];

    for (int gi = 0; gi < 8; ++gi) {
        const int row = i0 + gi + half * 8;
        float d = qsq[row * NH + h] + ksv - 2.0f * acc2[gi];
        d = fmaxf(d, 0.0f);
        const float bias = pb[((long long)row * LSEQ + col) * NH + h];
        float lg = w_l * (0.25f * acc1[gi] + bias - 0.5f * g * w_c * d);
        const float pm = mask[row] * mj;
        attn[(long long)h * LSEQ * LSEQ + (long long)row * LSEQ + col] =
            (pm == 0.0f) ? -1e9f : lg;
    }
}

// In-place row softmax over length-L rows (grid = NH*LSEQ rows, 256 threads)
__global__ void softmax_rows(float* __restrict__ attn) {
    __shared__ float red[256];
    float* p = attn + (long long)blockIdx.x * LSEQ;
    const int tid = threadIdx.x;

    float m = -1e30f;
    for (int j = tid; j < LSEQ; j += 256) m = fmaxf(m, p[j]);
    red[tid] = m; __syncthreads();
    for (int s = 128; s > 0; s >>= 1) {
        if (tid < s) red[tid] = fmaxf(red[tid], red[tid + s]);
        __syncthreads();
    }
    m = red[0]; __syncthreads();

    float sum = 0.f;
    for (int j = tid; j < LSEQ; j += 256) { float e = expf(p[j] - m); p[j] = e; sum += e; }
    red[tid] = sum; __syncthreads();
    for (int s = 128; s > 0; s >>= 1) {
        if (tid < s) red[tid] += red[tid + s];
        __syncthreads();
    }
    const float inv = 1.0f / red[0];
    for (int j = tid; j < LSEQ; j += 256) p[j] *= inv;
}

// pt_local = R^T (pt_global - t); write point_out + point_norms slots of s_cat
// one thread per (l, h, p);  ptg layout [l, h*PVN*3 + p*3 + c]
__global__ void point_finalize(const float* __restrict__ ptg,
                               const float* __restrict__ R, const float* __restrict__ t,
                               float* __restrict__ scat) {
    const int idx = blockIdx.x * blockDim.x + threadIdx.x;
    if (idx >= LSEQ * NH * PVN) return;
    const int l = idx >> 6;          // /(NH*PVN)
    const int rem = idx & 63;        // h*PVN + p
    const float* gp = ptg + (long long)l * (NH*PVN*3) + rem * 3;
    const float* Rl = R + l * 9;
    const float* tl = t + l * 3;
    const float x = gp[0] - tl[0], y = gp[1] - tl[1], z = gp[2] - tl[2];
    const float lx = Rl[0]*x + Rl[3]*y + Rl[6]*z;   // R^T
    const float ly = Rl[1]*x + Rl[4]*y + Rl[7]*z;
    const float lz = Rl[2]*x + Rl[5]*y + Rl[8]*z;
    float* o = scat + (long long)l * OUTD + (NH*CHD) + rem * 3;
    o[0] = lx; o[1] = ly; o[2] = lz;
    scat[(long long)l * OUTD + (NH*CHD + NH*PVN*3) + rem] =
        sqrtf(lx*lx + ly*ly + lz*lz + 1e-8f);
}

// out = (out + bo) * mask ; one thread per element of [L, CS]
__global__ void bias_mask(float* __restrict__ out, const float* __restrict__ bo,
                          const float* __restrict__ mask) {
    const int idx = blockIdx.x * blockDim.x + threadIdx.x;
    if (idx >= LSEQ * CS) return;
    const int l = idx >> 8, c = idx & 255;
    out[idx] = (out[idx] + bo[c]) * mask[l];
}

// -----------------------------------------------------------------------------
extern "C" void kernel_launch(void* const* d_in, const int* in_sizes, int n_in,
                              void* d_out, int out_size, void* d_ws, size_t ws_size,
                              hipStream_t stream) {
    const float* s    = (const float*)d_in[0];
    const float* z    = (const float*)d_in[1];
    const float* R    = (const float*)d_in[2];
    const float* t    = (const float*)d_in[3];
    const float* mask = (const float*)d_in[4];
    const float* Wq   = (const float*)d_in[5];
    const float* Wk   = (const float*)d_in[6];
    const float* Wv   = (const float*)d_in[7];
    const float* Wpb  = (const float*)d_in[8];
    const float* Wqp  = (const float*)d_in[9];
    const float* Wkp  = (const float*)d_in[10];
    const float* Wvp  = (const float*)d_in[11];
    const float* pw   = (const float*)d_in[12];
    const float* Wo   = (const float*)d_in[13];
    const float* bo   = (const float*)d_in[14];

    float* out_s = (float*)d_out;                 // [768,256]
    float* attn  = out_s + LSEQ * CS;             // [8,768,768]

    // workspace layout (floats)
    float* w    = (float*)d_ws;
    float* q    = w;                    // 768*128
    float* kbuf = q    + 98304;         // 768*128
    float* v    = kbuf + 98304;         // 768*128
    float* qp   = v    + 98304;         // 768*96   (local -> global in place)
    float* kp   = qp   + 73728;         // 768*96
    float* vp   = kp   + 73728;         // 768*192
    float* qsq  = vp   + 147456;        // 768*8
    float* ksq  = qsq  + 6144;          // 768*8
    float* gam  = ksq  + 6144;          // 8 (pad 32)
    float* pb   = gam  + 32;            // 589824*8  (pair bias, [ij, h])
    float* ptg  = pb   + 4718592;       // 768*192   (attn-weighted global pts)
    float* scat = ptg  + 147456;        // 768*1408  (concat before Wo)

    const dim3 wv(32);

    // ---- projections: s[768,256] @ W  (16x32 tiles) ----
    gemm_t<CS, 128, 128, CS, true><<<dim3(48, 4, 1), wv, 0, stream>>>(
        s, 0, Wq,  0, q,    0, LSEQ, 128);
    gemm_t<CS, 128, 128, CS, true><<<dim3(48, 4, 1), wv, 0, stream>>>(
        s, 0, Wk,  0, kbuf, 0, LSEQ, 128);
    gemm_t<CS, 128, 128, CS, true><<<dim3(48, 4, 1), wv, 0, stream>>>(
        s, 0, Wv,  0, v,    0, LSEQ, 128);
    gemm_t<CS, 96, 96, CS, true><<<dim3(48, 3, 1), wv, 0, stream>>>(
        s, 0, Wqp, 0, qp,   0, LSEQ, 96);
    gemm_t<CS, 96, 96, CS, true><<<dim3(48, 3, 1), wv, 0, stream>>>(
        s, 0, Wkp, 0, kp,   0, LSEQ, 96);
    gemm_t<CS, 192, 192, CS, true><<<dim3(48, 6, 1), wv, 0, stream>>>(
        s, 0, Wvp, 0, vp,   0, LSEQ, 192);

    // ---- frames: local -> global points, squared norms; gamma ----
    transform_pts<<<24, 256, 0, stream>>>(R, t, qp, qsq, PQK);
    transform_pts<<<24, 256, 0, stream>>>(R, t, kp, ksq, PQK);
    transform_pts<<<24, 256, 0, stream>>>(R, t, vp, nullptr, PVN);
    gamma_kernel<<<1, 32, 0, stream>>>(pw, gam);

    // ---- pair bias: z as [589824,128] @ Wpb[128,8] -> pb[ij, h] ----
    gemm_t<CZ, NH, NH, CZ, false><<<dim3(LSEQ * LSEQ / 16, 1, 1), wv, 0, stream>>>(
        z, 0, Wpb, 0, pb, 0, LSEQ * LSEQ, NH);

    // ---- fused logits (scalar + pair + point distance) -> attn region ----
    logits_kernel<<<dim3(48, 48, NH), wv, 0, stream>>>(q, kbuf, qp, kp, qsq, ksq,
                                                       pb, gam, mask, attn);

    // ---- softmax over j, in place ----
    softmax_rows<<<NH * LSEQ, 256, 0, stream>>>(attn);

    // ---- scalar_out: per-h attn[768,768] @ v[:,h,:](768,16) -> scat cols [h*16 .. ) ----
    gemm_t<LSEQ, NH * CHD, OUTD, LSEQ, false><<<dim3(48, 1, NH), wv, 0, stream>>>(
        attn, (long long)LSEQ * LSEQ,
        v,    CHD,
        scat, CHD,
        LSEQ, CHD);

    // ---- point out (global): per-h attn @ v_pts[:,h,:](768,24) -> ptg ----
    gemm_t<LSEQ, NH * PVN * 3, NH * PVN * 3, LSEQ, false><<<dim3(48, 2, NH), wv, 0, stream>>>(
        attn, (long long)LSEQ * LSEQ,
        vp,   PVN * 3,
        ptg,  PVN * 3,
        LSEQ, PVN * 3);

    // ---- pair_out: per-i attn[:, i, :](8,768) @ z[i](768,128) -> scat cols [384 .. ) ----
    gemm_t<LSEQ * LSEQ, CZ, CZ, LSEQ, true><<<dim3(1, 4, LSEQ), wv, 0, stream>>>(
        attn, LSEQ,
        z,    (long long)LSEQ * CZ,
        scat + (NH*CHD + NH*PVN*3 + NH*PVN), OUTD,
        NH, CZ);

    // ---- invert frames, norms, fill scat point slots ----
    point_finalize<<<192, 256, 0, stream>>>(ptg, R, t, scat);

    // ---- final projection: scat[768,1408] @ Wo[1408,256] -> s_update ----
    gemm_t<OUTD, CS, CS, OUTD, true><<<dim3(48, 8, 1), wv, 0, stream>>>(
        scat, 0, Wo, 0, out_s, 0, LSEQ, CS);

    bias_mask<<<LSEQ, 256, 0, stream>>>(out_s, bo, mask);
}